// TokenReductionLayer_4870492914225
// MI455X (gfx1250) — compile-verified
//
#include <hip/hip_runtime.h>

typedef float v2f __attribute__((ext_vector_type(2)));
typedef float v4f __attribute__((ext_vector_type(4)));
typedef float v8f __attribute__((ext_vector_type(8)));

#define TRL_B 4
#define TRL_S 8192
#define TRL_D 1024
#define TRL_R 4096

#define NTOK  (TRL_B * TRL_S)            // 32768 flattened tokens
#define NKEEP (TRL_B * TRL_R)            // 16384 kept rows (output rows)
#define NRED  (TRL_B * (TRL_S - TRL_R))  // 16384 reduced rows

// GEMM tiling
#define NT    128   // N columns per block
#define KC    64    // K chunk staged in LDS
#define WAVES 4     // waves per block
#define MT    2     // 16-row M tiles per wave (32 rows/wave) -> 2 WMMA per B frag
#define LDSS  68    // padded row stride (floats) -> conflict-free ds_load_b64

// ---------------------------------------------------------------------------
// Kernel 1: zero the flag array + compaction counter (ws re-inited every call).
__global__ void trl_init(int* __restrict__ flags, int* __restrict__ cnt) {
  int i = blockIdx.x * blockDim.x + threadIdx.x;
  if (i < NTOK) flags[i] = 0;
  if (i == 0) *cnt = 0;
}

// Kernel 2: flags[p] = 1 iff token p is a reduced token.
__global__ void trl_mark(const int* __restrict__ red, int* __restrict__ flags) {
  int i = blockIdx.x * blockDim.x + threadIdx.x;
  if (i < NRED) flags[red[i]] = 1;
}

// Kernel 3: compact output rows needing a GEMM contribution:
// out[j] += W @ x[keep[j]-1]  iff  keep[j]-1 is a reduced token.
// Store both the output row j and the A-source row (keep[j]-1).
__global__ void trl_compact(const int* __restrict__ keep,
                            const int* __restrict__ flags,
                            int* __restrict__ cnt, int* __restrict__ rows,
                            int* __restrict__ srcs) {
  int j = blockIdx.x * blockDim.x + threadIdx.x;
  if (j < NKEEP) {
    int k = keep[j];
    if (k > 0 && flags[k - 1]) {
      int p = atomicAdd(cnt, 1);
      rows[p] = j;
      srcs[p] = k - 1;
    }
  }
}

// Kernel 4: out[j] = x[keep[j]]   (vectorized float4 gather-copy, 134 MB)
__global__ void trl_copy(const float* __restrict__ x,
                         const int* __restrict__ keep,
                         float* __restrict__ out) {
  long i = (long)blockIdx.x * blockDim.x + threadIdx.x;  // float4 index
  if (i >= (long)NKEEP * (TRL_D / 4)) return;
  int j = (int)(i >> 8);          // D/4 == 256 float4 per row (const per block)
  int c = (int)(i & 255);
  long k = keep[j];
  ((v4f*)out)[i] = ((const v4f*)x)[k * (TRL_D / 4) + c];
}

// Kernel 5: out[j][nbase:nbase+NT] += x[srcs[...]] @ W^T for compacted rows.
// Block = 4 waves x (32 rows x 128 cols); W^T K-chunks staged in LDS.
// A-row indices are clamped (never zeroed): rows M >= nrows only feed C rows
// that are skipped at writeback, so EXEC stays all-ones through every WMMA.
__global__ __launch_bounds__(WAVES * 32)
void trl_gemm_add(const float* __restrict__ x, const float* __restrict__ W,
                  const int* __restrict__ rows, const int* __restrict__ srcs,
                  const int* __restrict__ cnt, float* __restrict__ out) {
  __shared__ float Bs[NT * LDSS];  // Bs[nl*LDSS + kk] = W[nbase+nl][kc+kk]

  const int tid   = threadIdx.x;
  const int lane  = tid & 31;
  const int wave  = tid >> 5;
  const int m     = lane & 15;   // M row (A) / N col (B,C) within tile
  const int g     = lane >> 4;   // K-group select for A/B fragments
  const int nbase = blockIdx.y * NT;
  const int tile0 = blockIdx.x * (WAVES * MT);

  const int count = *cnt;
  if (tile0 * 16 >= count) return;          // uniform whole-block exit (count>=1 past here)

  const int rowBase = (tile0 + wave * MT) * 16;   // first of this wave's 32 rows
  const int nrows0  = count - rowBase;            // valid rows in tile 0 (may be <=0)
  const int nrows1  = nrows0 - 16;                // valid rows in tile 1

  // Clamped per-lane A source pointers (always in-bounds, no EXEC divergence).
  const int last = count - 1;
  int i0 = rowBase + m;       i0 = i0 < last ? i0 : last;
  int i1 = rowBase + 16 + m;  i1 = i1 < last ? i1 : last;
  const float* aPtr0 = x + (long)srcs[i0] * TRL_D;
  const float* aPtr1 = x + (long)srcs[i1] * TRL_D;

  v8f acc0[NT / 16], acc1[NT / 16];
#pragma unroll
  for (int nt = 0; nt < NT / 16; ++nt) { acc0[nt] = (v8f){}; acc1[nt] = (v8f){}; }

  for (int kc = 0; kc < TRL_D; kc += KC) {
    __syncthreads();
    // Stage W^T chunk: thread t copies W[nbase+t][kc .. kc+KC) -> Bs[t][*]
    {
      const float* wp = W + (long)(nbase + tid) * TRL_D + kc;
      float* bp = &Bs[tid * LDSS];
#pragma unroll
      for (int q = 0; q < KC / 4; ++q)
        *(v4f*)(bp + 4 * q) = *(const v4f*)(wp + 4 * q);
    }
    __syncthreads();

    if (nrows0 > 0) {                        // wave-uniform: EXEC all-1s at WMMA
#pragma unroll
      for (int ks = 0; ks < KC; ks += 4) {
        v2f a0 = *(const v2f*)(aPtr0 + kc + ks + 2 * g);
        v2f a1 = *(const v2f*)(aPtr1 + kc + ks + 2 * g);
#pragma unroll
        for (int nt = 0; nt < NT / 16; ++nt) {
          v2f b = *(const v2f*)(&Bs[(nt * 16 + m) * LDSS + ks + 2 * g]);
          acc0[nt] = __builtin_amdgcn_wmma_f32_16x16x4_f32(
              false, a0, false, b, (short)0, acc0[nt], false, false);
          acc1[nt] = __builtin_amdgcn_wmma_f32_16x16x4_f32(
              false, a1, false, b, (short)0, acc1[nt], false, false);
        }
      }
    }
  }

  if (nrows0 <= 0) return;
  // Writeback: component v of acc holds (M = v + 8*g, N = nbase + nt*16 + m)
#pragma unroll
  for (int v = 0; v < 8; ++v) {
    int M = v + 8 * g;
    if (M < nrows0) {
      int j = rows[rowBase + M];
      float* op = out + (long)j * TRL_D + nbase + m;
#pragma unroll
      for (int nt = 0; nt < NT / 16; ++nt)
        op[nt * 16] += acc0[nt][v];
    }
    if (M < nrows1) {
      int j = rows[rowBase + 16 + M];
      float* op = out + (long)j * TRL_D + nbase + m;
#pragma unroll
      for (int nt = 0; nt < NT / 16; ++nt)
        op[nt * 16] += acc1[nt][v];
    }
  }
}

// ---------------------------------------------------------------------------
extern "C" void kernel_launch(void* const* d_in, const int* in_sizes, int n_in,
                              void* d_out, int out_size, void* d_ws, size_t ws_size,
                              hipStream_t stream) {
  const float* x    = (const float*)d_in[0];
  const float* W    = (const float*)d_in[1];
  const int*   keep = (const int*)d_in[2];
  const int*   red  = (const int*)d_in[3];
  float*       out  = (float*)d_out;

  // workspace layout
  int* flags = (int*)d_ws;                    // [NTOK]
  int* cnt   = flags + NTOK;                  // [1]
  int* rows  = flags + NTOK + 16;             // [NKEEP]
  int* srcs  = flags + NTOK + 16 + NKEEP;     // [NKEEP]

  trl_init<<<(NTOK + 255) / 256, 256, 0, stream>>>(flags, cnt);
  trl_mark<<<(NRED + 255) / 256, 256, 0, stream>>>(red, flags);
  trl_compact<<<(NKEEP + 255) / 256, 256, 0, stream>>>(keep, flags, cnt, rows, srcs);

  long n4 = (long)NKEEP * (TRL_D / 4);
  trl_copy<<<(unsigned)((n4 + 255) / 256), 256, 0, stream>>>(x, keep, out);

  dim3 grid((NKEEP / 16) / (WAVES * MT), TRL_D / NT);   // (128, 8)
  trl_gemm_add<<<grid, WAVES * 32, 0, stream>>>(x, W, rows, srcs, cnt, out);
}